// GCN_layer_69252052680936
// MI455X (gfx1250) — compile-verified
//
#include <hip/hip_runtime.h>

typedef __attribute__((ext_vector_type(2))) float v2f;
typedef __attribute__((ext_vector_type(4))) float v4f;
typedef __attribute__((ext_vector_type(8))) float v8f;

#define D 128   // D_IN == D_OUT == 128

// ---------------------------------------------------------------------------
// Phase 1: x[n][o] = sum_d input[n][d] * W[o][d] + b[o]   (fp32 WMMA 16x16x4)
// One block = 256 threads = 8 waves; block handles rows [16*bid, 16*bid+16),
// wave w computes the 16x16 tile at columns [16w, 16w+16).
// ---------------------------------------------------------------------------
__global__ __launch_bounds__(256) void gemm_wmma_f32(
    const float* __restrict__ A,      // [M, 128] node features
    const float* __restrict__ W,      // [128, 128] row-major (o, d)
    const float* __restrict__ bias,   // [128]
    float* __restrict__ X,            // [M, 128] projected output (workspace)
    int M)
{
    const int wave = threadIdx.x >> 5;      // 0..7 -> N tile
    const int lane = threadIdx.x & 31;
    const int half = lane >> 4;             // 0: lanes 0-15, 1: lanes 16-31
    const int l16  = lane & 15;

    const int m0 = blockIdx.x * 16;
    const int n  = wave * 16 + l16;         // global output column (B/D lane)

    // A row for this lane (same for both lane halves); clamp for tail safety.
    int arow = m0 + l16;
    if (arow >= M) arow = M - 1;
    const float* __restrict__ ap = A + (size_t)arow * D;
    const float* __restrict__ wp = W + (size_t)n * D;   // B[k][n] = W[n][k]

    v8f acc = {};
    #pragma unroll
    for (int k = 0; k < D; k += 4) {
        const int kk = k + 2 * half;        // lanes 16-31 carry K = {2,3} of slice
        v2f a; a.x = ap[kk]; a.y = ap[kk + 1];
        v2f b; b.x = wp[kk]; b.y = wp[kk + 1];
        acc = __builtin_amdgcn_wmma_f32_16x16x4_f32(
            /*neg_a=*/false, a, /*neg_b=*/false, b,
            /*c_mod=*/(short)0, acc, /*reuse_a=*/false, /*reuse_b=*/false);
    }

    const float bv = bias[n];
    #pragma unroll
    for (int i = 0; i < 8; ++i) {
        const int row = m0 + i + half * 8;  // D layout: VGPR i -> M = i (+8 hi half)
        if (row < M)
            X[(size_t)row * D + n] = acc[i] + bv;
    }
}

// ---------------------------------------------------------------------------
// Zero-fill d_out (poisoned by harness) with b128 stores.
// ---------------------------------------------------------------------------
__global__ __launch_bounds__(256) void zero_f32x4(float* __restrict__ out, int n4)
{
    const int i = blockIdx.x * blockDim.x + threadIdx.x;
    if (i < n4) {
        v4f z = {0.f, 0.f, 0.f, 0.f};
        ((v4f*)out)[i] = z;
    }
}

// ---------------------------------------------------------------------------
// Phase 2: out[dst] += val * x[src].  One wave per edge; lane handles a
// float4 slice (128 floats = 32 lanes * 4).  x lives in the 192MB L2, so the
// random gather and the fp32 atomic scatter are L2-resident.
// ---------------------------------------------------------------------------
__global__ __launch_bounds__(256) void spmm_scatter(
    const int*   __restrict__ esrc,
    const int*   __restrict__ edst,
    const float* __restrict__ eval,
    const float* __restrict__ X,
    float*       __restrict__ out,
    int E)
{
    const int wave = threadIdx.x >> 5;
    const int lane = threadIdx.x & 31;
    const int e = blockIdx.x * 8 + wave;
    if (e >= E) return;

    const int   s = esrc[e];
    const int   d = edst[e];
    const float v = eval[e];

    const v4f xv = ((const v4f*)(X + (size_t)s * D))[lane];   // global_load_b128
    float* __restrict__ op = out + (size_t)d * D + lane * 4;

    unsafeAtomicAdd(op + 0, v * xv.x);   // global_atomic_add_f32
    unsafeAtomicAdd(op + 1, v * xv.y);
    unsafeAtomicAdd(op + 2, v * xv.z);
    unsafeAtomicAdd(op + 3, v * xv.w);
}

// ---------------------------------------------------------------------------
// Entry point.
// Inputs: 0 layer_input[f32 M*128], 1 edge_src[i32 E], 2 edge_dst[i32 E],
//         3 edge_vals[f32 E], 4 W[f32 128*128], 5 b[f32 128]
// ---------------------------------------------------------------------------
extern "C" void kernel_launch(void* const* d_in, const int* in_sizes, int n_in,
                              void* d_out, int out_size, void* d_ws, size_t ws_size,
                              hipStream_t stream)
{
    const float* A    = (const float*)d_in[0];
    const int*   esrc = (const int*)  d_in[1];
    const int*   edst = (const int*)  d_in[2];
    const float* ev   = (const float*)d_in[3];
    const float* W    = (const float*)d_in[4];
    const float* bias = (const float*)d_in[5];
    float*       out  = (float*)d_out;
    float*       X    = (float*)d_ws;          // [M,128] fp32 intermediate (25.6 MB)

    const int M = in_sizes[0] / D;             // 50000
    const int E = in_sizes[1];                 // 800000

    const int mblocks = (M + 15) / 16;
    gemm_wmma_f32<<<mblocks, 256, 0, stream>>>(A, W, bias, X, M);

    const int n4 = out_size / 4;
    zero_f32x4<<<(n4 + 255) / 256, 256, 0, stream>>>(out, n4);

    spmm_scatter<<<(E + 7) / 8, 256, 0, stream>>>(esrc, edst, ev, X, out, E);
}